// CMOS_60181081752266
// MI455X (gfx1250) — compile-verified
//
#include <hip/hip_runtime.h>

// gfx1250 WMMA fragment types: 16x16x4 F32 -> A/B = 2 f32/lane, C/D = 8 f32/lane (wave32)
typedef float v2f __attribute__((ext_vector_type(2)));
typedef float v8f __attribute__((ext_vector_type(8)));

#define CLASSES          10
#define GRID_N           8192
#define FRAME            273
#define BLOCKS_PER_CLASS 35
#define WAVES_PER_BLOCK  8
#define WAVES_PER_CLASS  (BLOCKS_PER_CLASS * WAVES_PER_BLOCK) /* 280 >= 273 rows */

// Phase 1: one wave per patch row. |x| fragments are reduced on the matrix pipe:
// D = A x Ones + C  accumulates row-sums; every column of C is bitwise identical,
// so the wave total = (per-lane sum of 8 C VGPRs) + shfl_xor(.,16).
__global__ __launch_bounds__(256) void CMOS_frame_abssum_partial(
    const float* __restrict__ X, float* __restrict__ ws) {
  const int cls  = blockIdx.y;
  const int lane = threadIdx.x & 31;
  // Force wave-uniform (SGPR) row id so the branch below is a scalar branch and
  // EXEC stays all-1s around the WMMAs (ISA 7.12 requirement).
  const int wave = __builtin_amdgcn_readfirstlane(
      (int)(blockIdx.x * WAVES_PER_BLOCK) + (int)(threadIdx.x >> 5));

  const int s0 = (cls & 3) * 2048 + 888;  // first-dim start  (cls % 4)
  const int s1 = (cls >> 2) * 2048 + 888; // second-dim start (cls / 4)

  v8f c = {0.f, 0.f, 0.f, 0.f, 0.f, 0.f, 0.f, 0.f};
  const v2f ones = {1.0f, 1.0f};

  if (wave < FRAME) {
    const float* rp = X + (size_t)(s0 + wave) * GRID_N + s1; // 16B-aligned
    // Columns 0..255: two 128-wide chunks, float4 (global_load_b128) per lane.
#pragma unroll
    for (int k = 0; k < 2; ++k) {
      const float4 v = *reinterpret_cast<const float4*>(rp + k * 128 + lane * 4);
      v2f a0 = { fabsf(v.x), fabsf(v.y) };
      v2f a1 = { fabsf(v.z), fabsf(v.w) };
      c = __builtin_amdgcn_wmma_f32_16x16x4_f32(false, a0, false, ones,
                                                (short)0, c, false, false);
      c = __builtin_amdgcn_wmma_f32_16x16x4_f32(false, a1, false, ones,
                                                (short)0, c, false, false);
    }
    // Tail columns 256..272 (17 elems): zero-pad into A, WMMA stays full-EXEC.
    float t = (lane < (FRAME - 256)) ? fabsf(rp[256 + lane]) : 0.0f;
    v2f at = { t, 0.0f };
    c = __builtin_amdgcn_wmma_f32_16x16x4_f32(false, at, false, ones,
                                              (short)0, c, false, false);
  }

  float s = c[0] + c[1] + c[2] + c[3] + c[4] + c[5] + c[6] + c[7];
  s += __shfl_xor(s, 16, 32);  // pair lane L with L+16 -> full column sum == total
  if (lane == 0) ws[cls * WAVES_PER_CLASS + wave] = s;
}

// Phase 2: fixed-order (deterministic) tree reduction of the 280 partials per class.
__global__ __launch_bounds__(256) void CMOS_frame_abssum_final(
    const float* __restrict__ ws, float* __restrict__ out) {
  __shared__ float red[256];
  const int cls = blockIdx.x;
  const int t   = threadIdx.x;
  float s = ws[cls * WAVES_PER_CLASS + t];                     // t < 256 < 280
  if (t + 256 < WAVES_PER_CLASS) s += ws[cls * WAVES_PER_CLASS + t + 256];
  red[t] = s;
  __syncthreads();
#pragma unroll
  for (int off = 128; off >= 1; off >>= 1) {
    if (t < off) red[t] += red[t + off];
    __syncthreads();
  }
  if (t == 0) out[cls] = red[0];
}

extern "C" void kernel_launch(void* const* d_in, const int* in_sizes, int n_in,
                              void* d_out, int out_size, void* d_ws, size_t ws_size,
                              hipStream_t stream) {
  (void)in_sizes; (void)n_in; (void)out_size; (void)ws_size;
  const float* X   = (const float*)d_in[0];
  float*       out = (float*)d_out;
  float*       ws  = (float*)d_ws;  // uses 10*280*4 = 11,200 bytes

  dim3 grid1(BLOCKS_PER_CLASS, CLASSES);
  CMOS_frame_abssum_partial<<<grid1, 256, 0, stream>>>(X, ws);
  CMOS_frame_abssum_final<<<CLASSES, 256, 0, stream>>>(ws, out);
}